// BertCrf_46746424050031
// MI455X (gfx1250) — compile-verified
//
#include <hip/hip_runtime.h>
#include <hip/hip_bf16.h>
#include <math.h>

typedef float v2f __attribute__((ext_vector_type(2)));
typedef float v8f __attribute__((ext_vector_type(8)));

#define BDIM 64
#define SDIM 512
#define HDIM 768
#define TDIM 3
#define KSTEPS (HDIM / 4)          // 192 WMMA k-steps
#define EM_BYTES ((size_t)BDIM * SDIM * TDIM * sizeof(float))  // 384 KB

// ---------------------------------------------------------------------------
// Kernel 0: precompute the WMMA B-fragments of zero-padded W (768x3 -> 768x16)
// once. Fragment layout for V_WMMA_F32_16X16X4_F32 B (4x16):
//   lane l: n = l&15, rows K = {0,1} for l<16, {2,3} for l>=16.
// bfrag[it*32 + l] = float2{ Wp[it*4+koff][n], Wp[it*4+koff+1][n] }.
// All 2048 GEMM tiles share these 48 KB (stays L2/WGP$ resident).
// ---------------------------------------------------------------------------
__global__ __launch_bounds__(256)
void build_bfrag_kernel(const float* __restrict__ W, v2f* __restrict__ bfrag) {
  const int total = KSTEPS * 32;
  for (int idx = blockIdx.x * blockDim.x + threadIdx.x; idx < total;
       idx += gridDim.x * blockDim.x) {
    const int it   = idx >> 5;
    const int lane = idx & 31;
    const int n    = lane & 15;
    const int koff = (lane >> 4) << 1;
    const int k    = it * 4 + koff;
    v2f bf;
    bf.x = (n < TDIM) ? W[(k)     * TDIM + n] : 0.0f;
    bf.y = (n < TDIM) ? W[(k + 1) * TDIM + n] : 0.0f;
    bfrag[idx] = bf;
  }
}

// ---------------------------------------------------------------------------
// Kernel 1: emissions = hidden @ W + b via V_WMMA_F32_16X16X4_F32.
// One wave per 16-row tile of the (B*S, H) activation matrix. HBM-bound:
// streams 96 MB of `hidden` exactly once (~4.1 us at 23.3 TB/s). Inner loop
// is 2 x global_load_b64 + 1 wmma, fully unpredicated.
// ---------------------------------------------------------------------------
__global__ __launch_bounds__(32)
void emissions_wmma_kernel(const float* __restrict__ hidden,
                           const v2f*  __restrict__ bfrag,
                           const float* __restrict__ bias,
                           float* __restrict__ em) {
  const int tile = blockIdx.x;          // 16-row tile index
  const int lane = threadIdx.x;         // 0..31 (wave32)
  const int n    = lane & 15;
  const int hi   = lane >> 4;           // 0: lanes 0-15, 1: lanes 16-31
  const int koff = hi << 1;             // A rows K={0,1} vs K={2,3}

  // A fragment: M = lane%16, K advancing; lane reads 8B/iter from its row.
  const v2f* arow = (const v2f*)(hidden + (size_t)(tile * 16 + n) * HDIM + koff);
  const v2f* bptr = bfrag + lane;

  v8f c = {};  // 16x16 f32 accumulator
#pragma unroll 8
  for (int it = 0; it < KSTEPS; ++it) {
    v2f a = __builtin_nontemporal_load(arow + it * 2);  // hidden: stream, NT
    v2f b = bptr[it * 32];                              // shared B fragment
    // 8 args: (neg_a, A, neg_b, B, c_mod, C, reuse_a, reuse_b)
    c = __builtin_amdgcn_wmma_f32_16x16x4_f32(false, a, false, b,
                                              (short)0, c, false, false);
  }

  if (n < TDIM) {
    const float bn   = bias[n];
    const int  rbase = tile * 16 + (hi << 3);  // C/D: VGPR v -> M=v (+8 upper)
#pragma unroll
    for (int v = 0; v < 8; ++v) {
      em[(size_t)(rbase + v) * TDIM + n] = c[v] + bn;
    }
  }
}

// ---------------------------------------------------------------------------
// Kernel 2: zero the scalar output (graph-capture safe, no hipMemset).
// ---------------------------------------------------------------------------
__global__ void zero_out_kernel(float* __restrict__ out) {
  if (threadIdx.x == 0 && blockIdx.x == 0) out[0] = 0.0f;
}

// ---------------------------------------------------------------------------
// Kernel 3: CRF log-likelihood. One wave per batch element.
// Lane layout: l = j*4 + i (j,i in 0..2 active). The 3x3 logsumexp step is
// computed across lanes and reduced with shfl_xor within groups of 4.
// ---------------------------------------------------------------------------
__global__ __launch_bounds__(32)
void crf_kernel(const float* __restrict__ em,           // (B*S, 3)
                const float* __restrict__ start_trans,  // (3,)
                const float* __restrict__ end_trans,    // (3,)
                const float* __restrict__ trans,        // (3,3)
                const int*   __restrict__ mask,         // (B,S)
                const int*   __restrict__ tags,         // (B,S)
                float* __restrict__ out) {
  const int b = blockIdx.x;
  const int l = threadIdx.x;
  const float* E = em   + (size_t)b * SDIM * TDIM;
  const int*   M = mask + (size_t)b * SDIM;
  const int*   G = tags + (size_t)b * SDIM;

  const int  i   = l & 3;
  const int  j   = l >> 2;
  const bool act = (i < TDIM) && (j < TDIM);
  const float NEG = -3.0e38f;  // finite "-inf": avoids inf-inf = NaN

  const float tij = act ? trans[i * TDIM + j] : NEG;

  // alpha[i], replicated across all j-groups (valid on lanes with i<3)
  float alpha = (i < TDIM) ? (start_trans[i] + E[i]) : NEG;

  for (int s = 1; s < SDIM; ++s) {
    const float emj = act ? E[s * TDIM + j] : NEG;
    float x = act ? (alpha + tij + emj) : NEG;
    // max over i within the j-group (4 lanes)
    float m = fmaxf(x, __shfl_xor(x, 1, 4));
    m = fmaxf(m, __shfl_xor(m, 2, 4));
    float e = __expf(x - m);                 // inactive lanes contribute 0
    float ssum = e + __shfl_xor(e, 1, 4);
    ssum += __shfl_xor(ssum, 2, 4);
    const float nxt = m + __logf(ssum);      // nxt_j, same in all 4 group lanes
    // gather alpha_i for next step: group (l&3) holds nxt_{l&3} at lane (l&3)*4
    const float anew = __shfl(nxt, (l & 3) * 4);
    const bool  upd  = (M[s] > 0);
    alpha = upd ? ((i < TDIM) ? anew : NEG) : alpha;
  }

  // denominator: logsumexp_i(alpha_i + end_trans[i]); lane 0 holds the result
  float y  = (i < TDIM) ? (alpha + end_trans[i]) : NEG;
  float dm = fmaxf(y, __shfl_xor(y, 1, 4));
  dm = fmaxf(dm, __shfl_xor(dm, 2, 4));
  float de = __expf(y - dm);
  float dsum = de + __shfl_xor(de, 1, 4);
  dsum += __shfl_xor(dsum, 2, 4);
  const float den = dm + __logf(dsum);

  // numerator: lane-parallel over s, then wave reduction
  float num  = 0.0f;
  int   msum = 0;
  for (int s = l; s < SDIM; s += 32) {
    msum += M[s];
    if (s == 0) {
      const int t0 = G[0];
      num += start_trans[t0] + E[t0];
    } else {
      const float mk = (float)M[s];
      num += (trans[G[s - 1] * TDIM + G[s]] + E[s * TDIM + G[s]]) * mk;
    }
  }
#pragma unroll
  for (int off = 16; off >= 1; off >>= 1) {
    num  += __shfl_xor(num, off);
    msum += __shfl_xor(msum, off);
  }

  if (l == 0) {
    num += end_trans[G[msum - 1]];
    atomicAdd(out, num - den);
  }
}

// ---------------------------------------------------------------------------
extern "C" void kernel_launch(void* const* d_in, const int* in_sizes, int n_in,
                              void* d_out, int out_size, void* d_ws, size_t ws_size,
                              hipStream_t stream) {
  const float* hidden      = (const float*)d_in[0];  // (B,S,H)
  const float* W           = (const float*)d_in[1];  // (H,T)
  const float* bias        = (const float*)d_in[2];  // (T,)
  const float* start_trans = (const float*)d_in[3];  // (T,)
  const float* end_trans   = (const float*)d_in[4];  // (T,)
  const float* transitions = (const float*)d_in[5];  // (T,T)
  const int*   attn_mask   = (const int*)d_in[6];    // (B,S)
  const int*   tags        = (const int*)d_in[7];    // (B,S)
  float*       out         = (float*)d_out;          // scalar
  float*       em          = (float*)d_ws;                       // 384 KB
  v2f*         bfrag       = (v2f*)((char*)d_ws + EM_BYTES);     // +48 KB

  build_bfrag_kernel<<<24, 256, 0, stream>>>(W, bfrag);
  const int tiles = (BDIM * SDIM) / 16;  // 2048 waves, 16 rows each
  emissions_wmma_kernel<<<tiles, 32, 0, stream>>>(hidden, bfrag, bias, em);
  zero_out_kernel<<<1, 32, 0, stream>>>(out);
  crf_kernel<<<BDIM, 32, 0, stream>>>(em, start_trans, end_trans, transitions,
                                      attn_mask, tags, out);
}